// Spline_27736898798308
// MI455X (gfx1250) — compile-verified
//
#include <hip/hip_runtime.h>
#include <hip/hip_bf16.h>
#include <stdint.h>

// ---------------------------------------------------------------------------
// Spline evaluation, MI455X (gfx1250).
// Memory-bound (~137 MB traffic -> ~5.9us floor @ 23.3 TB/s, ~0.26 GFLOP).
// Uses gfx1250 async global->LDS instructions (ASYNCcnt path) for input
// staging and LDS-transposed b128 stores for the dominant output stream.
// ---------------------------------------------------------------------------

#define K_INTERVALS 16
#define N_SAMPLES   64
#define CPB         4     // curves per block (4 groups of 64 lanes)

// gfx1250 async global->LDS copy, 64-bit VGPR address form (saddr = off).
__device__ __forceinline__ void async_ld_b32(unsigned lds_off, const void* g) {
  asm volatile("global_load_async_to_lds_b32 %0, %1, off"
               :: "v"(lds_off), "v"((unsigned long long)(uintptr_t)g)
               : "memory");
}
__device__ __forceinline__ void async_ld_b128(unsigned lds_off, const void* g) {
  asm volatile("global_load_async_to_lds_b128 %0, %1, off"
               :: "v"(lds_off), "v"((unsigned long long)(uintptr_t)g)
               : "memory");
}
__device__ __forceinline__ void wait_asynccnt0() {
  asm volatile("s_wait_asynccnt 0" ::: "memory");
}
// Low 32 bits of a generic pointer to LDS == byte offset in the LDS window.
__device__ __forceinline__ unsigned lds_off(const void* p) {
  return (unsigned)(uintptr_t)p;
}

__global__ __launch_bounds__(256)
void Spline_27736898798308_kernel(const float*  __restrict__ t_in,     // (N)
                                  const float*  __restrict__ J_in,     // (C,17,3)
                                  const float*  __restrict__ CP_in,    // (C,32,3)
                                  const int*    __restrict__ mask_in,  // (C,17)
                                  const float*  __restrict__ W_in,     // (C,16)
                                  const int*    __restrict__ ipc_in,   // (C)
                                  float*        __restrict__ out,      // (C,64,3)
                                  int C) {
  __shared__ float sJ[CPB][52];                       // 51 used
  __shared__ __align__(16) float sCP[CPB][96];
  __shared__ float sFirst[CPB][48];
  __shared__ float sW[CPB][K_INTERVALS];
  __shared__ float sCum[CPB][K_INTERVALS];
  __shared__ int   sMask[CPB][18];                    // 17 used
  __shared__ int   sIpc[CPB];
  __shared__ float sT[N_SAMPLES];
  __shared__ __align__(16) float sOut[CPB * N_SAMPLES * 3];   // 768 floats

  const int tid  = threadIdx.x;
  const int g    = tid >> 6;          // curve-group within block, 0..3
  const int lane = tid & 63;          // 0..63 (one t-sample per lane)
  const int c    = blockIdx.x * CPB + g;
  const bool cOK = (c < C);

  // ---- Phase 1: stage inputs into LDS (async tensor path for the bulk) ----
  if (tid < N_SAMPLES) sT[tid] = t_in[tid];

  if (cOK) {
    const float* Jg  = J_in  + (size_t)c * 51;
    const float* CPg = CP_in + (size_t)c * 96;

    // J: 51 dwords, per-curve base only 4B-aligned -> async b32 per lane.
    if (lane < 51) async_ld_b32(lds_off(&sJ[g][lane]), Jg + lane);
    // CP: 96 floats = 24 x 16B, per-curve base 384B-aligned -> async b128.
    if (lane < 24) async_ld_b128(lds_off(&sCP[g][lane * 4]), CPg + lane * 4);

    // Small per-curve metadata via regular loads.
    if (lane < K_INTERVALS) sW[g][lane] = W_in[(size_t)c * K_INTERVALS + lane];
    if (lane < 17)          sMask[g][lane] = mask_in[(size_t)c * 17 + lane];
    if (lane == 0)          sIpc[g] = ipc_in[c];

    // Warm L2 for the next block-stride of control points (192MB L2 holds
    // the entire 60MB input set). -> global_prefetch_b8
    if (lane < 24) {
      size_t cn = (size_t)c + (size_t)CPB * gridDim.x / gridDim.x * CPB; // c + 16
      (void)cn;
      if (c + CPB < C)
        __builtin_prefetch(CP_in + (size_t)(c + CPB) * 96 + lane * 4, 0, 3);
    }
  }
  wait_asynccnt0();       // drain this wave's ASYNCcnt before the barrier
  __syncthreads();

  // ---- Phase 2: per-interval derived data (lanes 0..15 of each group) ----
  if (cOK && lane < K_INTERVALS) {
    const int k = lane;
    float cum = 0.f;
    #pragma unroll
    for (int j = 0; j < K_INTERVALS; ++j)
      if (j <= k) cum += sW[g][j];
    sCum[g][k] = cum;

    const int ipcv = sIpc[g];
    const bool derive = (k >= 1) & (k < ipcv) &
                        (sMask[g][k] != 0) & (sMask[g][k + 1] != 0);
    #pragma unroll
    for (int d = 0; d < 3; ++d) {
      // raw[k][j] lives at sCP[(k*2 + j)*3 + d]
      float fv;
      if (derive) fv = 2.f * sJ[g][k * 3 + d] - sCP[g][((k - 1) * 2 + 1) * 3 + d];
      else        fv = sCP[g][(k * 2) * 3 + d];
      sFirst[g][k * 3 + d] = fv;
    }
  }
  __syncthreads();

  // ---- Phase 3: evaluate one t-sample per lane ----
  if (cOK) {
    const float tv = sT[lane];
    float acc = 0.f;
    int idx = 0;
    #pragma unroll
    for (int k = 0; k < K_INTERVALS - 1; ++k) {   // cum[:-1]
      acc += sW[g][k];
      idx += (tv >= acc) ? 1 : 0;
    }
    const int ipcv = sIpc[g];
    idx = min(idx, ipcv - 1);

    const float w  = sW[g][idx];
    const float s  = sCum[g][idx] - w;
    float lt = (tv - s) / fmaxf(w, 1e-12f);
    lt = fminf(fmaxf(lt, 0.f), 1.f);
    const float omt = 1.f - lt;
    const float b0 = omt * omt * omt;
    const float b1 = 3.f * lt * omt * omt;
    const float b2 = 3.f * lt * lt * omt;
    const float b3 = lt * lt * lt;

    #pragma unroll
    for (int d = 0; d < 3; ++d) {
      float r = b0 * sJ[g][idx * 3 + d];                       // P0 = J[idx]
      r = fmaf(b1, sFirst[g][idx * 3 + d], r);                 // P1 = first[idx]
      r = fmaf(b2, sCP[g][(idx * 2 + 1) * 3 + d], r);          // P2 = raw[idx][1]
      r = fmaf(b3, sJ[g][(idx + 1) * 3 + d], r);               // P3 = J[idx+1]
      sOut[g * (N_SAMPLES * 3) + lane * 3 + d] = r;            // stride-3: bank-clean
    }
  }
  __syncthreads();

  // ---- Phase 4: dense b128 flush of the 768-float block tile ----
  const int nv4 = CPB * N_SAMPLES * 3 / 4;   // 192 float4
  if (tid < nv4) {
    const int cLocal = (tid * 4) / (N_SAMPLES * 3);
    if (blockIdx.x * CPB + cLocal < C) {
      float4* o4 = reinterpret_cast<float4*>(out + (size_t)blockIdx.x * (CPB * N_SAMPLES * 3));
      o4[tid] = reinterpret_cast<const float4*>(sOut)[tid];
    }
  }
}

extern "C" void kernel_launch(void* const* d_in, const int* in_sizes, int n_in,
                              void* d_out, int out_size, void* d_ws, size_t ws_size,
                              hipStream_t stream) {
  const float* t    = (const float*)d_in[0];
  const float* J    = (const float*)d_in[1];
  const float* CP   = (const float*)d_in[2];
  const int*   mask = (const int*)d_in[3];
  const float* W    = (const float*)d_in[4];
  const int*   ipc  = (const int*)d_in[5];
  float*       out  = (float*)d_out;

  const int C = in_sizes[5];               // intervals_per_curve is (C,)
  const int blocks = (C + CPB - 1) / CPB;

  Spline_27736898798308_kernel<<<blocks, 256, 0, stream>>>(
      t, J, CP, mask, W, ipc, out, C);
}